// MLP_mia_white_weight_sim2_65300682768667
// MI455X (gfx1250) — compile-verified
//
#include <hip/hip_runtime.h>
#include <hip/hip_bf16.h>

// ---------------------------------------------------------------------------
// Problem constants (from reference): T=8, N=100000, D=64, E=262144,
// HIDDEN=[64,32].  Pipeline:
//   1) per-node L2 norms of z1/z2            (memory-bound, coalesced)
//   2) per-edge weighted cosine sims -> sim  (memory-bound random gather)
//   3) MLP over E rows using V_WMMA_F32_16X16X4_F32 (fp32-exact matrix path)
// ---------------------------------------------------------------------------

#define TT   8
#define NN   100000
#define DD   64
#define EE   262144
#define H1   64
#define H2   32
#define EPSF 1e-8f

typedef float v2f __attribute__((ext_vector_type(2)));
typedef float v8f __attribute__((ext_vector_type(8)));

// fp32 WMMA: D(16x16) = A(16x4) * B(4x16) + C.  8-arg VOP3P form.
__device__ __forceinline__ v8f wmma_f32_16x16x4(v2f a, v2f b, v8f c) {
    return __builtin_amdgcn_wmma_f32_16x16x4_f32(
        /*neg_a=*/false, a, /*neg_b=*/false, b,
        /*c_mod=*/(short)0, c, /*reuse_a=*/false, /*reuse_b=*/false);
}

// Butterfly sum across a 16-lane group (masks < 16 stay inside the group).
__device__ __forceinline__ float sum16(float v) {
    v += __shfl_xor(v, 1, 32);
    v += __shfl_xor(v, 2, 32);
    v += __shfl_xor(v, 4, 32);
    v += __shfl_xor(v, 8, 32);
    return v;
}

// ---------------------------------------------------------------------------
// Kernel 1: per-(t,node) vector norms for z1 and z2.
// 16 lanes cooperate on one 64-float vector -> one fully coalesced 256B read.
// ---------------------------------------------------------------------------
__global__ void __launch_bounds__(256)
norms_kernel(const float* __restrict__ z1, const float* __restrict__ z2,
             float* __restrict__ n1, float* __restrict__ n2) {
    const int gid = blockIdx.x * blockDim.x + threadIdx.x;
    const int vec = gid >> 4;
    const int sub = gid & 15;
    const int per = TT * NN;
    if (vec >= 2 * per) return;

    const bool second = vec >= per;
    const int v = second ? vec - per : vec;
    const float* __restrict__ z = second ? z2 : z1;

    const float4 a = ((const float4*)(z + (size_t)v * DD))[sub];
    float s = a.x * a.x + a.y * a.y + a.z * a.z + a.w * a.w;
    s = sum16(s);
    if (sub == 0) (second ? n2 : n1)[v] = sqrtf(s);
}

// ---------------------------------------------------------------------------
// Kernel 2: per-edge weighted cosine similarities.
// 16 lanes per edge; each t: 4 coalesced float4 gathers + butterfly dot.
// sim[e][t] = w1[t]*cos1 + w2[t]*cos2, stored row-major (E x 8).
// ---------------------------------------------------------------------------
__global__ void __launch_bounds__(256)
sim_kernel(const int* __restrict__ edges,
           const float* __restrict__ z1, const float* __restrict__ z2,
           const float* __restrict__ n1, const float* __restrict__ n2,
           const float* __restrict__ w1, const float* __restrict__ w2,
           float* __restrict__ sim) {
    const int gid = blockIdx.x * blockDim.x + threadIdx.x;
    const int e   = gid >> 4;
    const int sub = gid & 15;
    if (e >= EE) return;

    const int src = edges[2 * e];
    const int dst = edges[2 * e + 1];

    float mine = 0.0f;  // lane `sub` keeps sim[e][sub] (sub < 8)

#pragma unroll
    for (int t = 0; t < TT; ++t) {
        const int   ns = t * NN + src;
        const int   nd = t * NN + dst;
        const size_t is = (size_t)ns * DD;
        const size_t id = (size_t)nd * DD;

        const float4 a1 = ((const float4*)(z1 + is))[sub];
        const float4 b1 = ((const float4*)(z1 + id))[sub];
        const float4 a2 = ((const float4*)(z2 + is))[sub];
        const float4 b2 = ((const float4*)(z2 + id))[sub];

        float d1 = a1.x * b1.x + a1.y * b1.y + a1.z * b1.z + a1.w * b1.w;
        float d2 = a2.x * b2.x + a2.y * b2.y + a2.z * b2.z + a2.w * b2.w;
        d1 = sum16(d1);
        d2 = sum16(d2);

        const float c1 = d1 / fmaxf(n1[ns] * n1[nd], EPSF);
        const float c2 = d2 / fmaxf(n2[ns] * n2[nd], EPSF);
        const float s  = w1[t] * c1 + w2[t] * c2;
        if (sub == t) mine = s;
    }
    if (sub < TT) sim[(size_t)e * TT + sub] = mine;  // coalesced 32B/group
}

// ---------------------------------------------------------------------------
// Kernel 3: MLP with V_WMMA_F32_16X16X4_F32.
// One wave per 16-row tile; 8 waves (256 thr) per block.
//
// A layout (16x4 fp32, 2 VGPRs = v2f): lane L holds M = L%16,
//   K = 2*(L/16) + {0,1}  -> float2 load from a row-major A tile.
// B layout (4x16 fp32, mirrored): lane L holds N = L%16,
//   K = 2*(L/16) + {0,1}  -> float2 load from row-major W (fan_out x fan_in),
//   since B = W^T means B[k][n] = W[n][k].
// C/D layout: VGPR r, lane L -> (M = r + 8*(L/16), N = L%16).
//
// h1 is restaged via LDS (row stride 66 floats: conflict-free and 8B-aligned
// for the float2 A reads of layer 2).
// ---------------------------------------------------------------------------
#define LDS_STRIDE 66

__global__ void __launch_bounds__(256)
mlp_kernel(const float* __restrict__ sim,
           const float* __restrict__ W1, const float* __restrict__ b1,
           const float* __restrict__ W2, const float* __restrict__ b2,
           const float* __restrict__ Wp, const float* __restrict__ bp,
           float* __restrict__ out) {
    __shared__ float lds[8][16 * LDS_STRIDE];

    const int wave = threadIdx.x >> 5;
    const int lane = threadIdx.x & 31;
    const int half = lane >> 4;   // 0: K even pair low, 1: high
    const int sub  = lane & 15;   // M (for A) or N (for B)

    const size_t tile = (size_t)blockIdx.x * 8 + wave;   // E/16 tiles total
    const size_t row0 = tile * 16;
    float* __restrict__ h = &lds[wave][0];

    // ---- Layer 1: (16x8) @ (8x64) + b1, relu ------------------------------
    const float* simrow = sim + (row0 + sub) * TT;
    const v2f a0 = *(const v2f*)(simrow + 2 * half);       // K 0..3
    const v2f a1 = *(const v2f*)(simrow + 4 + 2 * half);   // K 4..7

#pragma unroll
    for (int nt = 0; nt < 4; ++nt) {
        const int n0 = nt * 16;
        const float* w1row = W1 + (size_t)(n0 + sub) * TT; // W1[n][k], k<8
        const v2f bB0 = *(const v2f*)(w1row + 2 * half);
        const v2f bB1 = *(const v2f*)(w1row + 4 + 2 * half);

        const float bias = b1[n0 + sub];
        v8f c = {bias, bias, bias, bias, bias, bias, bias, bias};
        c = wmma_f32_16x16x4(a0, bB0, c);
        c = wmma_f32_16x16x4(a1, bB1, c);

#pragma unroll
        for (int r = 0; r < 8; ++r) {
            const int M = r + 8 * half;
            h[M * LDS_STRIDE + n0 + sub] = fmaxf(c[r], 0.0f);
        }
    }
    __syncthreads();

    // ---- Layer 2: (16x64) @ (64x32) + b2, relu ----------------------------
    v8f d0, d1;
    {
        const float bias0 = b2[sub];
        const float bias1 = b2[16 + sub];
        d0 = (v8f){bias0, bias0, bias0, bias0, bias0, bias0, bias0, bias0};
        d1 = (v8f){bias1, bias1, bias1, bias1, bias1, bias1, bias1, bias1};
    }
#pragma unroll
    for (int ks = 0; ks < 16; ++ks) {
        const int k0 = ks * 4 + 2 * half;                    // even -> 8B aligned
        const v2f a = *(const v2f*)(h + sub * LDS_STRIDE + k0);

        const v2f bB0 = *(const v2f*)(W2 + (size_t)sub * H1 + k0);
        const v2f bB1 = *(const v2f*)(W2 + (size_t)(16 + sub) * H1 + k0);
        d0 = wmma_f32_16x16x4(a, bB0, d0);
        d1 = wmma_f32_16x16x4(a, bB1, d1);
    }

    // ---- Layer 3: (16x32) @ (32x1) + bp  (VALU + 16-lane reduction) -------
    const float wlo = Wp[sub];
    const float whi = Wp[16 + sub];
    const float bpv = bp[0];

#pragma unroll
    for (int r = 0; r < 8; ++r) {
        float p = fmaxf(d0[r], 0.0f) * wlo + fmaxf(d1[r], 0.0f) * whi;
        p = sum16(p);
        if (sub == 0) out[row0 + 8 * half + r] = p + bpv;
    }
}

// ---------------------------------------------------------------------------
// Launcher.  Inputs (setup_inputs order):
//  0 edge_list (E,2) int  | 1 z1 (T,N,D) f32 | 2 z2 (T,N,D) f32
//  3 weight_vec1 (1,T)    | 4 weight_vec2 (1,T)
//  5 W1 (64,8) | 6 b1 (64,) | 7 W2 (32,64) | 8 b2 (32,) | 9 Wp (1,32) | 10 bp (1,)
// Workspace: n1 [T*N] f32 | n2 [T*N] f32 | sim [E*8] f32  (~14.8 MB)
// ---------------------------------------------------------------------------
extern "C" void kernel_launch(void* const* d_in, const int* in_sizes, int n_in,
                              void* d_out, int out_size, void* d_ws, size_t ws_size,
                              hipStream_t stream) {
    const int*   edges = (const int*)  d_in[0];
    const float* z1    = (const float*)d_in[1];
    const float* z2    = (const float*)d_in[2];
    const float* w1    = (const float*)d_in[3];
    const float* w2    = (const float*)d_in[4];
    const float* W1    = (const float*)d_in[5];
    const float* b1    = (const float*)d_in[6];
    const float* W2    = (const float*)d_in[7];
    const float* b2    = (const float*)d_in[8];
    const float* Wp    = (const float*)d_in[9];
    const float* bp    = (const float*)d_in[10];
    float*       outp  = (float*)d_out;

    float* n1  = (float*)d_ws;
    float* n2  = n1 + (size_t)TT * NN;
    float* sim = n2 + (size_t)TT * NN;

    // Kernel 1: 2*T*N vectors, 16 lanes each.
    {
        const long long threads = 2LL * TT * NN * 16;
        const int blocks = (int)((threads + 255) / 256);
        norms_kernel<<<blocks, 256, 0, stream>>>(z1, z2, n1, n2);
    }
    // Kernel 2: E edges, 16 lanes each.
    {
        const long long threads = (long long)EE * 16;
        const int blocks = (int)((threads + 255) / 256);
        sim_kernel<<<blocks, 256, 0, stream>>>(edges, z1, z2, n1, n2, w1, w2, sim);
    }
    // Kernel 3: E/16 tiles, 8 tiles (waves) per block.
    {
        const int tiles  = EE / 16;     // 16384
        const int blocks = tiles / 8;   // 2048
        mlp_kernel<<<blocks, 256, 0, stream>>>(sim, W1, b1, W2, b2, Wp, bp, outp);
    }
}